// Han_69295002354093
// MI455X (gfx1250) — compile-verified
//
#include <hip/hip_runtime.h>
#include <hip/hip_bf16.h>
#include <math.h>

// ---------------------------------------------------------------------------
// HAN forward for MI455X (gfx1250), fp32 WMMA (V_WMMA_F32_16X16X4_F32).
// Latency-bound recurrent workload: gather + input-projection fused into the
// GRU step kernel; attention projection GEMM fused with tanh / dot-v /
// cross-lane reduction. One wave (32 threads) per block; M-blocked tiles
// (MT row-tiles per wave) amortize B-fragment traffic MT-fold.
// __launch_bounds__(32,1) gives each wave the full 256-VGPR budget so the
// MT-blocked accumulators (up to 128 VGPRs) never spill to scratch.
// ---------------------------------------------------------------------------

typedef __attribute__((ext_vector_type(2))) float v2f;
typedef __attribute__((ext_vector_type(8))) float v8f;

static __device__ __forceinline__ v8f wmma_f32(v2f a, v2f b, v8f c) {
  // args: (neg_a, A, neg_b, B, c_mod, C, reuse_a, reuse_b)
  return __builtin_amdgcn_wmma_f32_16x16x4_f32(false, a, false, b, (short)0, c,
                                               false, false);
}

static __device__ __forceinline__ float sigmoidf_(float x) {
  return 1.0f / (1.0f + __expf(-x));
}

// Problem constants (match reference)
#define BB 64
#define SS 16
#define WW 32
#define EE 300
#define HH 256   // HW == HS
#define NC 10
#define NRW (BB * SS)  // 1024 rows at word level
#define NRS (BB)       // 64 rows at sentence level
#define DD (2 * HH)    // 512

// ---------------------------------------------------------------------------
// zero helper
// ---------------------------------------------------------------------------
__global__ void k_zero(float* p, long n) {
  long i = (long)blockIdx.x * blockDim.x + threadIdx.x;
  if (i < n) p[i] = 0.0f;
}

// ---------------------------------------------------------------------------
// One GRU timestep for one direction (blockIdx.z == dir).
// blockDim = 32 (one wave). Grid: (Nr/(16*MT), HH/16, 2).
// Each wave computes MT row-tiles sharing the same B (weight) fragments:
//   gates = x_t @ Wih^T + bih  and  h @ Whh^T + bhh  via WMMA f32,
// then the GRU nonlinearity for its (16*MT rows x 16 h-columns) tile.
//   Word level:  x_t row m = emb[tok[m*tstride + t]], ld = EE
//   Sent level:  x_t row m = Xsrc[(m*tstride + t)*ld], ld = DD
// h layout per buffer: [2 dirs][Nr][HH]. y_out: [Nr][2*HH], dir d in cols
// d*HH..d*HH+HH (concatenation of the two directions).
// ---------------------------------------------------------------------------
template <int MT>
__global__ __launch_bounds__(32, 1) void k_gru_step(
    const float* __restrict__ Xsrc, const int* __restrict__ tok, int ld,
    int tstride, int t, int Ein, int Nr,
    const float* __restrict__ Wih,  // [2][3*HH][Ein]
    const float* __restrict__ Whh,  // [2][3*HH][HH]
    const float* __restrict__ bih,  // [2][3*HH]
    const float* __restrict__ bhh,  // [2][3*HH]
    const float* __restrict__ h_prev,  // [2][Nr][HH]
    float* __restrict__ h_next,        // [2][Nr][HH]
    float* __restrict__ y_out) {       // [Nr][2*HH]
  const int lane = threadIdx.x;
  const int m0 = blockIdx.x * (16 * MT);
  const int c0 = blockIdx.y * 16;
  const int d = blockIdx.z;

  const float* wih = Wih + (size_t)d * 3 * HH * Ein;
  const float* whh = Whh + (size_t)d * 3 * HH * HH;
  const float* bi = bih + d * 3 * HH;
  const float* bh = bhh + d * 3 * HH;
  const float* hp = h_prev + (size_t)d * Nr * HH;
  float* hn = h_next + (size_t)d * Nr * HH;

  const int koff = (lane >> 4) * 2;    // A/B fragment K offset for this lane
  const int bn = lane & 15;            // B-fragment column (N) for this lane

  // A-fragment row pointers per M-tile (gather fused for word level)
  const float* ax[MT];
  const float* ah[MT];
#pragma unroll
  for (int mt = 0; mt < MT; ++mt) {
    const int am = m0 + mt * 16 + (lane & 15);
    const int xrow = tok ? tok[(size_t)am * tstride + t] : (am * tstride + t);
    ax[mt] = Xsrc + (size_t)xrow * ld;
    ah[mt] = hp + (size_t)am * HH;
  }

  // B row pointers: B[k][n] = W[gate*HH + c0 + n][k]
  const float* wir = wih + (size_t)(0 * HH + c0 + bn) * Ein;
  const float* wiz = wih + (size_t)(1 * HH + c0 + bn) * Ein;
  const float* win = wih + (size_t)(2 * HH + c0 + bn) * Ein;
  const float* whr = whh + (size_t)(0 * HH + c0 + bn) * HH;
  const float* whz = whh + (size_t)(1 * HH + c0 + bn) * HH;
  const float* whn = whh + (size_t)(2 * HH + c0 + bn) * HH;

  v8f accR[MT];   // i_r + h_r
  v8f accZ[MT];   // i_z + h_z
  v8f accIN[MT];  // i_n (input part only — must stay separate)
  v8f accHN[MT];  // h_n (hidden part only)
#pragma unroll
  for (int mt = 0; mt < MT; ++mt) {
    accR[mt] = (v8f){};
    accZ[mt] = (v8f){};
    accIN[mt] = (v8f){};
    accHN[mt] = (v8f){};
  }

  // input projection: K = Ein (300 or 512), step 4; B shared across M-tiles
  for (int k = 0; k < Ein; k += 4) {
    v2f br, bz, bnv;
    br.x = wir[k + koff]; br.y = wir[k + koff + 1];
    bz.x = wiz[k + koff]; bz.y = wiz[k + koff + 1];
    bnv.x = win[k + koff]; bnv.y = win[k + koff + 1];
#pragma unroll
    for (int mt = 0; mt < MT; ++mt) {
      v2f a;
      a.x = ax[mt][k + koff];
      a.y = ax[mt][k + koff + 1];
      accR[mt] = wmma_f32(a, br, accR[mt]);
      accZ[mt] = wmma_f32(a, bz, accZ[mt]);
      accIN[mt] = wmma_f32(a, bnv, accIN[mt]);
    }
  }
  // hidden projection: K = HH (256), step 4
  for (int k = 0; k < HH; k += 4) {
    v2f br, bz, bnv;
    br.x = whr[k + koff]; br.y = whr[k + koff + 1];
    bz.x = whz[k + koff]; bz.y = whz[k + koff + 1];
    bnv.x = whn[k + koff]; bnv.y = whn[k + koff + 1];
#pragma unroll
    for (int mt = 0; mt < MT; ++mt) {
      v2f a;
      a.x = ah[mt][k + koff];
      a.y = ah[mt][k + koff + 1];
      accR[mt] = wmma_f32(a, br, accR[mt]);
      accZ[mt] = wmma_f32(a, bz, accZ[mt]);
      accHN[mt] = wmma_f32(a, bnv, accHN[mt]);
    }
  }

  // GRU gate math (C/D layout: M = r + 8*(lane/16), N = lane%16)
  const int c = c0 + (lane & 15);
  const float bir = bi[c], biz = bi[HH + c], bin = bi[2 * HH + c];
  const float bhr = bh[c], bhz = bh[HH + c], bhn = bh[2 * HH + c];
#pragma unroll
  for (int mt = 0; mt < MT; ++mt) {
    const int mbase = m0 + mt * 16 + (lane >> 4) * 8;
#pragma unroll
    for (int r = 0; r < 8; ++r) {
      const int m = mbase + r;
      const float hold = hp[(size_t)m * HH + c];
      const float rg = sigmoidf_(accR[mt][r] + bir + bhr);
      const float zg = sigmoidf_(accZ[mt][r] + biz + bhz);
      const float ng = tanhf(accIN[mt][r] + bin + rg * (accHN[mt][r] + bhn));
      const float h2 = (1.0f - zg) * ng + zg * hold;
      hn[(size_t)m * HH + c] = h2;
      y_out[(size_t)m * DD + d * HH + c] = h2;
    }
  }
}

// ---------------------------------------------------------------------------
// Attention scores: scores[row] = tanh(H[row,:] @ Wa^T + ba) . v
// H: [Mrows][DD] (row = t*Nr + n). WMMA GEMM tile with fused tanh, dot-v,
// shfl_xor reduction over the 16 column lanes, atomicAdd per row.
// MT row-tiles per wave share B (Wa) fragments.
// Grid: (Mrows/(16*MT), DD/16), blockDim = 32.
// ---------------------------------------------------------------------------
template <int MT>
__global__ __launch_bounds__(32, 1) void k_attn_scores(
    const float* __restrict__ H,
    const float* __restrict__ Wa,  // [DD][DD]
    const float* __restrict__ ba,  // [DD]
    const float* __restrict__ v,   // [DD]
    float* __restrict__ scores) {
  const int lane = threadIdx.x;
  const int m0 = blockIdx.x * (16 * MT);
  const int n0 = blockIdx.y * 16;
  const int koff = (lane >> 4) * 2;

  const float* arow[MT];
#pragma unroll
  for (int mt = 0; mt < MT; ++mt)
    arow[mt] = H + (size_t)(m0 + mt * 16 + (lane & 15)) * DD;
  const float* brow = Wa + (size_t)(n0 + (lane & 15)) * DD;

  v8f acc[MT];
#pragma unroll
  for (int mt = 0; mt < MT; ++mt) acc[mt] = (v8f){};

  for (int k = 0; k < DD; k += 4) {
    v2f b;
    b.x = brow[k + koff];
    b.y = brow[k + koff + 1];
#pragma unroll
    for (int mt = 0; mt < MT; ++mt) {
      v2f a;
      a.x = arow[mt][k + koff];
      a.y = arow[mt][k + koff + 1];
      acc[mt] = wmma_f32(a, b, acc[mt]);
    }
  }

  const int c = n0 + (lane & 15);
  const float bac = ba[c], vc = v[c];
#pragma unroll
  for (int mt = 0; mt < MT; ++mt) {
    float part[8];
#pragma unroll
    for (int r = 0; r < 8; ++r) part[r] = tanhf(acc[mt][r] + bac) * vc;
    // reduce over the 16 lanes sharing the same row group (lane>>4)
#pragma unroll
    for (int off = 8; off >= 1; off >>= 1) {
#pragma unroll
      for (int r = 0; r < 8; ++r) part[r] += __shfl_xor(part[r], off, 32);
    }
    if ((lane & 15) == 0) {
      const int mbase = m0 + mt * 16 + (lane >> 4) * 8;
#pragma unroll
      for (int r = 0; r < 8; ++r) atomicAdd(&scores[mbase + r], part[r]);
    }
  }
}

// ---------------------------------------------------------------------------
// Softmax over t + weighted sum: out[n][:] = sum_t softmax(scores[:,n])[t] * H[t*Nr+n][:]
// Grid: Nr blocks, 256 threads.
// ---------------------------------------------------------------------------
__global__ void k_attn_combine(const float* __restrict__ H,
                               const float* __restrict__ scores, int T, int Nr,
                               float* __restrict__ out) {
  const int n = blockIdx.x;
  const int tid = threadIdx.x;
  __shared__ float a[64];
  if (tid == 0) {
    float mx = -1e30f;
    for (int t = 0; t < T; ++t) mx = fmaxf(mx, scores[(size_t)t * Nr + n]);
    float s = 0.0f;
    for (int t = 0; t < T; ++t) {
      float e = __expf(scores[(size_t)t * Nr + n] - mx);
      a[t] = e;
      s += e;
    }
    float inv = 1.0f / s;
    for (int t = 0; t < T; ++t) a[t] *= inv;
  }
  __syncthreads();
  for (int dcol = tid; dcol < DD; dcol += blockDim.x) {
    float acc = 0.0f;
    for (int t = 0; t < T; ++t)
      acc += a[t] * H[((size_t)t * Nr + n) * DD + dcol];
    out[(size_t)n * DD + dcol] = acc;
  }
}

// ---------------------------------------------------------------------------
// Classifier + log_softmax. Grid: BB blocks, 256 threads.
// ---------------------------------------------------------------------------
__global__ void k_classifier(const float* __restrict__ doc,
                             const float* __restrict__ W,  // [NC][DD]
                             const float* __restrict__ b,  // [NC]
                             float* __restrict__ out) {
  const int bb = blockIdx.x;
  const int tid = threadIdx.x;
  __shared__ float red[256];
  __shared__ float logits[NC];
  const float* drow = doc + (size_t)bb * DD;
  for (int c = 0; c < NC; ++c) {
    float p = 0.0f;
    for (int d2 = tid; d2 < DD; d2 += 256) p += drow[d2] * W[(size_t)c * DD + d2];
    red[tid] = p;
    __syncthreads();
    for (int s = 128; s > 0; s >>= 1) {
      if (tid < s) red[tid] += red[tid + s];
      __syncthreads();
    }
    if (tid == 0) logits[c] = red[0] + b[c];
    __syncthreads();
  }
  if (tid == 0) {
    float mx = -1e30f;
    for (int c = 0; c < NC; ++c) mx = fmaxf(mx, logits[c]);
    float s = 0.0f;
    for (int c = 0; c < NC; ++c) s += __expf(logits[c] - mx);
    float lse = mx + __logf(s);
    for (int c = 0; c < NC; ++c) out[bb * NC + c] = logits[c] - lse;
  }
}

// ---------------------------------------------------------------------------
// Workspace layout (floats)
// ---------------------------------------------------------------------------
#define OFF_HW0 0L                                   // word h buf0: [2][NRW][HH]
#define SZ_HWB ((long)2 * NRW * HH)                  // 524288
#define OFF_HW1 (OFF_HW0 + SZ_HWB)                   // word h buf1
#define OFF_HS0 (OFF_HW1 + SZ_HWB)                   // sent h buf0: [2][NRS][HH]
#define SZ_HSB ((long)2 * NRS * HH)                  // 32768
#define OFF_HS1 (OFF_HS0 + SZ_HSB)
#define OFF_SCW (OFF_HS1 + SZ_HSB)                   // word scores [WW*NRW]
#define SZ_SCW ((long)WW * NRW)
#define OFF_SCS (OFF_SCW + SZ_SCW)                   // sent scores [SS*NRS]
#define SZ_SCS ((long)SS * NRS)
#define ZERO_PREFIX (OFF_SCS + SZ_SCS)               // everything above zeroed
#define OFF_HWALL (ZERO_PREFIX)                      // [WW][NRW][DD] = 16.7M
#define SZ_HWALL ((long)WW * NRW * DD)
#define OFF_SENT (OFF_HWALL + SZ_HWALL)              // [NRW][DD]
#define SZ_SENT ((long)NRW * DD)
#define OFF_HSALL (OFF_SENT + SZ_SENT)               // [SS][NRS][DD]
#define SZ_HSALL ((long)SS * NRS * DD)
#define OFF_DOC (OFF_HSALL + SZ_HSALL)               // [NRS][DD]
#define SZ_DOC ((long)NRS * DD)
// total ~19.1M floats (~76.5 MB) — fits typical scratch allocations

extern "C" void kernel_launch(void* const* d_in, const int* in_sizes, int n_in,
                              void* d_out, int out_size, void* d_ws,
                              size_t ws_size, hipStream_t stream) {
  const int* tokens = (const int*)d_in[0];     // [B,S,W]
  const float* emb = (const float*)d_in[1];    // [V,E]
  const float* w_Wih = (const float*)d_in[2];  // [2,3HW,E]
  const float* w_Whh = (const float*)d_in[3];  // [2,3HW,HW]
  const float* w_bih = (const float*)d_in[4];  // [2,3HW]
  const float* w_bhh = (const float*)d_in[5];  // [2,3HW]
  const float* wa_W = (const float*)d_in[6];   // [2HW,2HW]
  const float* wa_b = (const float*)d_in[7];   // [2HW]
  const float* wa_v = (const float*)d_in[8];   // [2HW]
  const float* s_Wih = (const float*)d_in[9];  // [2,3HS,2HW]
  const float* s_Whh = (const float*)d_in[10]; // [2,3HS,HS]
  const float* s_bih = (const float*)d_in[11];
  const float* s_bhh = (const float*)d_in[12];
  const float* sa_W = (const float*)d_in[13];
  const float* sa_b = (const float*)d_in[14];
  const float* sa_v = (const float*)d_in[15];
  const float* fc_W = (const float*)d_in[16];  // [NC,2HS]
  const float* fc_b = (const float*)d_in[17];  // [NC]
  float* out = (float*)d_out;                  // [B,NC]

  float* ws = (float*)d_ws;
  float* hw_buf[2] = {ws + OFF_HW0, ws + OFF_HW1};
  float* hs_buf[2] = {ws + OFF_HS0, ws + OFF_HS1};
  float* scores_w = ws + OFF_SCW;
  float* scores_s = ws + OFF_SCS;
  float* hw_all = ws + OFF_HWALL;
  float* sent = ws + OFF_SENT;
  float* hs_all = ws + OFF_HSALL;
  float* doc = ws + OFF_DOC;

  // zero h0 buffers + score accumulators
  {
    long n = ZERO_PREFIX;
    k_zero<<<dim3((unsigned)((n + 255) / 256)), dim3(256), 0, stream>>>(ws, n);
  }

  // ---- word-level bi-GRU: 32 timesteps, both directions per launch --------
  // MT=4: 512 waves/step, 4x less weight traffic per step.
  for (int t = 0; t < WW; ++t) {
    k_gru_step<4><<<dim3(NRW / 64, HH / 16, 2), dim3(32), 0, stream>>>(
        emb, tokens, EE, WW, t, EE, NRW, w_Wih, w_Whh, w_bih, w_bhh,
        hw_buf[t & 1], hw_buf[(t + 1) & 1], hw_all + (size_t)t * NRW * DD);
  }

  // ---- word attention -----------------------------------------------------
  // MT=8: Wa fragment traffic cut 8x; still 8192 waves.
  k_attn_scores<8><<<dim3((WW * NRW) / 128, DD / 16), dim3(32), 0, stream>>>(
      hw_all, wa_W, wa_b, wa_v, scores_w);
  k_attn_combine<<<dim3(NRW), dim3(256), 0, stream>>>(hw_all, scores_w, WW,
                                                      NRW, sent);

  // ---- sentence-level bi-GRU: 16 timesteps --------------------------------
  // MT=1: only 64 rows, keep parallelism (128 waves/step).
  for (int t = 0; t < SS; ++t) {
    k_gru_step<1><<<dim3(NRS / 16, HH / 16, 2), dim3(32), 0, stream>>>(
        sent, nullptr, DD, SS, t, DD, NRS, s_Wih, s_Whh, s_bih, s_bhh,
        hs_buf[t & 1], hs_buf[(t + 1) & 1], hs_all + (size_t)t * NRS * DD);
  }

  // ---- sentence attention -------------------------------------------------
  k_attn_scores<8><<<dim3((SS * NRS) / 128, DD / 16), dim3(32), 0, stream>>>(
      hs_all, sa_W, sa_b, sa_v, scores_s);
  k_attn_combine<<<dim3(NRS), dim3(256), 0, stream>>>(hs_all, scores_s, SS,
                                                      NRS, doc);

  // ---- classifier + log_softmax ------------------------------------------
  k_classifier<<<dim3(BB), dim3(256), 0, stream>>>(doc, fc_W, fc_b, out);
}